// VisualFeatureGraph_62715112457021
// MI455X (gfx1250) — compile-verified
//
#include <hip/hip_runtime.h>
#include <math.h>

typedef __attribute__((ext_vector_type(2))) float v2f;
typedef __attribute__((ext_vector_type(8))) float v8f;

#define N_CLS   256
#define FDIM    768
#define HDIM    384
#define NBATCH  128

// ---------------------------------------------------------------------------
// Generic f32 WMMA GEMM:  C[b] = A @ B[b]   (A shared across batch)
//   TRANSB=false : B is K x N row-major (ldb = row stride)
//   TRANSB=true  : B is N x K row-major (computes A @ B^T)
// Each wave computes a 64x16 tile: 4 accumulators share one B fragment.
// Requires M%64==0, N%16==0, K%4==0. Uses V_WMMA_F32_16X16X4_F32 (exact f32).
// ---------------------------------------------------------------------------
template<bool TRANSB>
__global__ __launch_bounds__(256)
void wmma_gemm_f32_kernel(const float* __restrict__ A, const float* __restrict__ B,
                          float* __restrict__ C,
                          int M, int N, int K, int lda, int ldb, int ldc,
                          long long strideB, long long strideC, int totalWaves)
{
    int tid  = blockIdx.x * blockDim.x + threadIdx.x;
    int wave = tid >> 5;
    if (wave >= totalWaves) return;           // wave-uniform guard (EXEC stays full)
    int lane = tid & 31;

    int nTilesN = N >> 4;
    int nTilesM = M >> 6;
    int nb   = wave % nTilesN;
    int rest = wave / nTilesN;
    int mb   = rest % nTilesM;
    int bat  = rest / nTilesM;

    const float* Bb = B + (long long)bat * strideB;
    float*       Cb = C + (long long)bat * strideC;

    int m0   = mb << 6;
    int n0   = nb << 4;
    int mr   = lane & 15;       // row (A) / col (B,C) index within tile
    int half = lane >> 4;       // K-pair selector for A/B, row-half for C/D

    v8f acc0 = {}, acc1 = {}, acc2 = {}, acc3 = {};

    for (int k = 0; k < K; k += 4) {
        int ka = k + 2 * half;
        v2f bv;
        if (TRANSB) {
            const float* bp = Bb + (long long)(n0 + mr) * ldb + ka;
            bv.x = bp[0]; bv.y = bp[1];
        } else {
            bv.x = Bb[(long long)ka * ldb + n0 + mr];
            bv.y = Bb[(long long)(ka + 1) * ldb + n0 + mr];
        }
        const float* ap = A + (long long)(m0 + mr) * lda + ka;
        v2f a0; a0.x = ap[0];            a0.y = ap[1];
        v2f a1; a1.x = ap[16 * lda];     a1.y = ap[16 * lda + 1];
        v2f a2; a2.x = ap[32 * lda];     a2.y = ap[32 * lda + 1];
        v2f a3; a3.x = ap[48 * lda];     a3.y = ap[48 * lda + 1];
        acc0 = __builtin_amdgcn_wmma_f32_16x16x4_f32(false, a0, false, bv, (short)0, acc0, false, false);
        acc1 = __builtin_amdgcn_wmma_f32_16x16x4_f32(false, a1, false, bv, (short)0, acc1, false, false);
        acc2 = __builtin_amdgcn_wmma_f32_16x16x4_f32(false, a2, false, bv, (short)0, acc2, false, false);
        acc3 = __builtin_amdgcn_wmma_f32_16x16x4_f32(false, a3, false, bv, (short)0, acc3, false, false);
    }

#pragma unroll
    for (int r = 0; r < 8; r++) {
        int row = r + 8 * half;
        Cb[(long long)(m0 +  0 + row) * ldc + n0 + mr] = acc0[r];
        Cb[(long long)(m0 + 16 + row) * ldc + n0 + mr] = acc1[r];
        Cb[(long long)(m0 + 32 + row) * ldc + n0 + mr] = acc2[r];
        Cb[(long long)(m0 + 48 + row) * ldc + n0 + mr] = acc3[r];
    }
}

// ---------------------------------------------------------------------------
// Row-wise L2 normalization of class_features (rows 0..255) and
// context_features (rows 256..511).
// ---------------------------------------------------------------------------
__global__ __launch_bounds__(256)
void rownorm_kernel(const float* __restrict__ cf, const float* __restrict__ ctx,
                    float* __restrict__ cfn, float* __restrict__ ctxn)
{
    __shared__ float sm[256];
    int r = blockIdx.x;
    const float* src = (r < N_CLS) ? cf  : ctx;
    float*       dst = (r < N_CLS) ? cfn : ctxn;
    int row = r & (N_CLS - 1);
    int t = threadIdx.x;

    float s = 0.f;
    for (int d = t; d < FDIM; d += 256) {
        float v = src[row * FDIM + d];
        s += v * v;
    }
    sm[t] = s;
    __syncthreads();
    for (int off = 128; off; off >>= 1) {
        if (t < off) sm[t] += sm[t + off];
        __syncthreads();
    }
    float invn = 1.0f / sqrtf(sm[0]);
    for (int d = t; d < FDIM; d += 256)
        dst[row * FDIM + d] = src[row * FDIM + d] * invn;
}

// ---------------------------------------------------------------------------
// sim_weight[i,j] = sigmoid( sum_k relu(a[i,k]-a[j,k]+b1[k]) * w2[k] + b2 )
// One wave per (i,j) pair; lane-strided over k=0..383, shuffle reduction.
// ---------------------------------------------------------------------------
__global__ __launch_bounds__(256)
void simweight_kernel(const float* __restrict__ amat, const float* __restrict__ b1,
                      const float* __restrict__ w2, const float* __restrict__ b2,
                      float* __restrict__ simw)
{
    int tid  = blockIdx.x * blockDim.x + threadIdx.x;
    int wave = tid >> 5;
    int lane = tid & 31;
    int i = wave >> 8;
    int j = wave & 255;

    float s = 0.f;
    for (int k = lane; k < HDIM; k += 32) {
        float h = amat[i * HDIM + k] - amat[j * HDIM + k] + b1[k];
        h = fmaxf(h, 0.f);
        s += h * w2[k];
    }
    for (int off = 16; off; off >>= 1) s += __shfl_xor(s, off, 32);
    if (lane == 0) {
        float z = s + b2[0];
        simw[wave] = 1.0f / (1.0f + expf(-z));
    }
}

// A = (co > 50) ? 1 : 0 ;  M = vs * A
__global__ __launch_bounds__(256)
void mask_kernel(const int* __restrict__ co, const float* __restrict__ vs,
                 float* __restrict__ amask, float* __restrict__ mmat)
{
    int idx = blockIdx.x * blockDim.x + threadIdx.x;
    float a = ((float)co[idx] > 50.0f) ? 1.0f : 0.0f;
    amask[idx] = a;
    mmat[idx]  = vs[idx] * a;
}

// Top-5 per row of co (descending, first-occurrence tie-break like lax.top_k).
__global__ __launch_bounds__(256)
void topk_kernel(const int* __restrict__ co, float* __restrict__ topv, int* __restrict__ topi)
{
    int i = blockIdx.x * blockDim.x + threadIdx.x;
    if (i >= N_CLS) return;
    int picked[5];
    for (int t = 0; t < 5; t++) {
        int best = -1, bv = -1;
        for (int j = 0; j < N_CLS; j++) {
            bool skip = false;
            for (int u = 0; u < t; u++) if (picked[u] == j) skip = true;
            if (skip) continue;
            int v = co[i * N_CLS + j];
            if (v > bv) { bv = v; best = j; }
        }
        picked[t] = best;
        topv[i * 5 + t] = (float)bv;
        topi[i * 5 + t] = best;
    }
}

// Assemble un-normalized E (tier1/tier2/tier3 blend, zero diagonal).
__global__ __launch_bounds__(256)
void assemble_kernel(const float* __restrict__ vs, const float* __restrict__ cs,
                     const float* __restrict__ simw, const float* __restrict__ ssum,
                     const float* __restrict__ scnt, const int* __restrict__ co,
                     const int* __restrict__ cnt, const float* __restrict__ topv,
                     const int* __restrict__ topi, float* __restrict__ Ep)
{
    int idx = blockIdx.x * blockDim.x + threadIdx.x;
    int i = idx >> 8, j = idx & 255;

    float vsij = vs[idx], csij = cs[idx];
    float combined = 0.7f * vsij + 0.3f * csij;

    float ci = (float)cnt[i], cj = (float)cnt[j];
    float ratio = fminf(ci, cj) / fmaxf(fmaxf(ci, cj), 1.0f);
    float e1 = combined * simw[idx] * ratio;

    float cof  = (float)co[idx];
    float conf = cof * (1.0f / 50.0f);
    float sc   = scnt[idx];
    float avg  = ssum[idx] / fmaxf(sc, 1.0f);
    float e2   = (sc > 0.0f) ? (conf * combined + (1.0f - conf) * avg) : (conf * combined);

    float trans = 0.0f, tot = 0.0f;
#pragma unroll
    for (int p = 0; p < 5; p++) {
        float tvi = topv[i * 5 + p];
        int   ii  = topi[i * 5 + p];
        bool  vi  = tvi > 10.0f;
        float vsin_i = vs[i * N_CLS + ii];
#pragma unroll
        for (int q = 0; q < 5; q++) {
            float tvj = topv[j * 5 + q];
            int   jj  = topi[j * 5 + q];
            float ccross = (float)co[ii * N_CLS + jj];
            bool  m  = vi && (tvj > 10.0f) && (ccross > 0.0f);
            float w  = m ? (tvi * tvj * (1.0f / 50.0f)) : 0.0f;
            float sim = vsin_i * vs[j * N_CLS + jj] * vs[ii * N_CLS + jj];
            trans += w * sim;
            tot   += w;
        }
    }
    float e3 = (tot > 0.0f) ? (trans / tot) : (0.1f * csij);

    bool t1 = cof > 50.0f;
    bool t2 = (cof > 10.0f) && !t1;
    float E = t1 ? e1 : (t2 ? e2 : e3);
    if (i == j) E = 0.0f;
    Ep[idx] = E;
}

// Block-wise max reduction (256 elems per block).
__global__ __launch_bounds__(256)
void reduce_max_kernel(const float* __restrict__ src, float* __restrict__ dst, int n)
{
    __shared__ float sm[256];
    int t = threadIdx.x;
    int idx = blockIdx.x * 256 + t;
    sm[t] = (idx < n) ? src[idx] : -3.402823e38f;
    __syncthreads();
    for (int off = 128; off; off >>= 1) {
        if (t < off) sm[t] = fmaxf(sm[t], sm[t + off]);
        __syncthreads();
    }
    if (t == 0) dst[blockIdx.x] = sm[0];
}

// E = (mx>0 ? E/mx : E) + I
__global__ __launch_bounds__(256)
void finalize_E_kernel(float* __restrict__ Ep, const float* __restrict__ mxp)
{
    int idx = blockIdx.x * blockDim.x + threadIdx.x;
    float mx = mxp[0];
    float v = Ep[idx];
    if (mx > 0.f) v /= mx;
    int i = idx >> 8, j = idx & 255;
    if (i == j) v += 1.0f;
    Ep[idx] = v;
}

// ---------------------------------------------------------------------------
template<bool TRANSB>
static inline void launch_gemm(const float* A, const float* B, float* C,
                               int M, int N, int K, int lda, int ldb, int ldc,
                               int nbatch, long long sB, long long sC, hipStream_t stream)
{
    int totalWaves = nbatch * (M >> 6) * (N >> 4);
    int blocks = (totalWaves + 7) / 8;
    wmma_gemm_f32_kernel<TRANSB><<<blocks, 256, 0, stream>>>(
        A, B, C, M, N, K, lda, ldb, ldc, sB, sC, totalWaves);
}

extern "C" void kernel_launch(void* const* d_in, const int* in_sizes, int n_in,
                              void* d_out, int out_size, void* d_ws, size_t ws_size,
                              hipStream_t stream)
{
    const float* x   = (const float*)d_in[0];   // (128,256,768)
    const float* cf  = (const float*)d_in[1];   // (256,768)
    const float* ctx = (const float*)d_in[2];   // (256,768)
    const float* w1  = (const float*)d_in[3];   // (768,384)
    const float* b1  = (const float*)d_in[4];   // (384,)
    const float* w2  = (const float*)d_in[5];   // (384,1)
    const float* b2  = (const float*)d_in[6];   // (1,)
    const int*   co  = (const int*)d_in[7];     // (256,256)
    const int*   cnt = (const int*)d_in[8];     // (256,)
    float* out = (float*)d_out;                 // (128,256,768)

    float* W     = (float*)d_ws;
    float* Ep    = W;                 // 65536
    float* vs    = W + 65536;         // 65536
    float* cs    = W + 131072;        // 65536
    float* simw  = W + 196608;        // 65536
    float* ssum  = W + 262144;        // 65536
    float* scnt  = W + 327680;        // 65536
    float* amask = W + 393216;        // 65536
    float* mmat  = W + 458752;        // 65536
    float* cfn   = W + 524288;        // 196608
    float* ctxn  = W + 720896;        // 196608
    float* amat  = W + 917504;        // 98304 (256x384)
    float* topv  = W + 1015808;       // 1280
    int*   topi  = (int*)(W + 1017088); // 1280
    float* red   = W + 1018368;       // 257

    // 1) L2-normalize feature rows
    rownorm_kernel<<<512, 256, 0, stream>>>(cf, ctx, cfn, ctxn);

    // 2) a = class_features @ w1         (256x768 @ 768x384)
    launch_gemm<false>(cf, w1, amat, N_CLS, HDIM, FDIM, FDIM, HDIM, HDIM, 1, 0, 0, stream);

    // 3) vs = cfn @ cfn^T ; cs = ctxn @ ctxn^T
    launch_gemm<true>(cfn,  cfn,  vs, N_CLS, N_CLS, FDIM, FDIM, FDIM, N_CLS, 1, 0, 0, stream);
    launch_gemm<true>(ctxn, ctxn, cs, N_CLS, N_CLS, FDIM, FDIM, FDIM, N_CLS, 1, 0, 0, stream);

    // 4) fused pairwise relu-dot sigmoid (65536 waves)
    simweight_kernel<<<8192, 256, 0, stream>>>(amat, b1, w2, b2, simw);

    // 5) masks, then shared_sum = M@M^T, shared_cnt = A@A^T
    mask_kernel<<<256, 256, 0, stream>>>(co, vs, amask, mmat);
    launch_gemm<true>(mmat,  mmat,  ssum, N_CLS, N_CLS, N_CLS, N_CLS, N_CLS, N_CLS, 1, 0, 0, stream);
    launch_gemm<true>(amask, amask, scnt, N_CLS, N_CLS, N_CLS, N_CLS, N_CLS, N_CLS, 1, 0, 0, stream);

    // 6) top-5 per row
    topk_kernel<<<1, 256, 0, stream>>>(co, topv, topi);

    // 7) assemble E, max-normalize, add identity
    assemble_kernel<<<256, 256, 0, stream>>>(vs, cs, simw, ssum, scnt, co, cnt, topv, topi, Ep);
    reduce_max_kernel<<<256, 256, 0, stream>>>(Ep, red, 65536);
    reduce_max_kernel<<<1, 256, 0, stream>>>(red, red + 256, 256);
    finalize_E_kernel<<<256, 256, 0, stream>>>(Ep, red + 256);

    // 8) out[b] = (I + E_norm) @ x[b]   -- the 12.9 GFLOP batched WMMA GEMM
    launch_gemm<false>(Ep, x, out, N_CLS, FDIM, N_CLS, N_CLS, FDIM, FDIM,
                       NBATCH, (long long)N_CLS * FDIM, (long long)N_CLS * FDIM, stream);
}